// GCN_54992761258609
// MI455X (gfx1250) — compile-verified
//
#include <hip/hip_runtime.h>
#include <hip/hip_bf16.h>
#include <math.h>

#define N_NODES 500000
#define N_EDGES 8000000
#define D 16
#define N_CLS 5

typedef __attribute__((ext_vector_type(2))) float v2f;
typedef __attribute__((ext_vector_type(8))) float v8f;

// ---------------------------------------------------------------------------
// 1) deg[i] = 1 (self loop)  then atomic += 1 per incoming edge
// ---------------------------------------------------------------------------
__global__ void init_deg_kernel(float* __restrict__ deg) {
    int i = blockIdx.x * blockDim.x + threadIdx.x;
    if (i < N_NODES) deg[i] = 1.0f;
}

__global__ void deg_accum_kernel(const long long* __restrict__ col,
                                 float* __restrict__ deg) {
    int e = blockIdx.x * blockDim.x + threadIdx.x;
    if (e < N_EDGES) atomicAdd(&deg[(int)col[e]], 1.0f);
}

__global__ void dinv_kernel(float* __restrict__ deg) {
    int i = blockIdx.x * blockDim.x + threadIdx.x;
    if (i < N_NODES) {
        float d = deg[i];
        deg[i] = (d > 0.0f) ? rsqrtf(d) : 0.0f;   // in-place deg -> dinv
    }
}

// ---------------------------------------------------------------------------
// 2) xw = x @ W_conv  via V_WMMA_F32_16X16X4_F32.  One wave per 16-row tile.
//    A (16x4 f32): lane<16 holds K=0,1; lane>=16 holds K=2,3 (2 VGPRs)
//    B (4x16 f32): mirrored striping; C/D: 8 VGPRs, M=v / M=v+8 halves.
// ---------------------------------------------------------------------------
__global__ void xw_wmma_kernel(const float* __restrict__ x,
                               const float* __restrict__ W,
                               float* __restrict__ xw) {
    int wave = (blockIdx.x * blockDim.x + threadIdx.x) >> 5;
    int lane = threadIdx.x & 31;
    if (wave >= N_NODES / 16) return;          // 31250 tiles, whole-wave exit

    int n   = lane & 15;                       // row of A / col of B
    int hi  = lane >> 4;                       // 0: K pair {0,1}, 1: {2,3}
    const float* xrow = x + (size_t)(wave * 16 + n) * D;

    v8f c = {};
#pragma unroll
    for (int kc = 0; kc < 4; ++kc) {
        int kb = kc * 4 + hi * 2;              // starting K for this lane half
        v2f a, b;
        a.x = xrow[kb];
        a.y = xrow[kb + 1];
        b.x = W[kb * D + n];
        b.y = W[(kb + 1) * D + n];
        // 8 args: (neg_a, A, neg_b, B, c_mod, C, reuse_a, reuse_b)
        c = __builtin_amdgcn_wmma_f32_16x16x4_f32(
                false, a, false, b, (short)0, c, false, false);
    }

    float* out = xw + (size_t)wave * 16 * D;
#pragma unroll
    for (int v = 0; v < 8; ++v) {
        int m = v + hi * 8;
        out[m * D + n] = c[v];
    }
}

// ---------------------------------------------------------------------------
// 3) accum[i] = xw[i] * dinv[i]^2   (self-loop message, no atomics needed)
// ---------------------------------------------------------------------------
__global__ void init_accum_kernel(const float* __restrict__ xw,
                                  const float* __restrict__ dinv,
                                  float* __restrict__ accum) {
    int i = blockIdx.x * blockDim.x + threadIdx.x;
    if (i >= N_NODES) return;
    float s = dinv[i];
    s = s * s;
    const float4* src = (const float4*)(xw + (size_t)i * D);
    float4* dst = (float4*)(accum + (size_t)i * D);
#pragma unroll
    for (int q = 0; q < 4; ++q) {
        float4 v = src[q];
        dst[q] = make_float4(v.x * s, v.y * s, v.z * s, v.w * s);
    }
}

// ---------------------------------------------------------------------------
// 4) per-edge gather + normalized scatter-add (16 f32 atomics per edge)
// ---------------------------------------------------------------------------
__global__ void scatter_kernel(const long long* __restrict__ row_idx,
                               const long long* __restrict__ col_idx,
                               const float* __restrict__ xw,
                               const float* __restrict__ dinv,
                               float* __restrict__ accum) {
    int e = blockIdx.x * blockDim.x + threadIdx.x;
    if (e >= N_EDGES) return;
    int r = (int)row_idx[e];
    int c = (int)col_idx[e];
    float norm = dinv[r] * dinv[c];
    const float* src = xw + (size_t)r * D;
    float* dst = accum + (size_t)c * D;
#pragma unroll
    for (int q = 0; q < 4; ++q) {
        float4 v = *(const float4*)(src + q * 4);
        atomicAdd(dst + q * 4 + 0, v.x * norm);
        atomicAdd(dst + q * 4 + 1, v.y * norm);
        atomicAdd(dst + q * 4 + 2, v.z * norm);
        atomicAdd(dst + q * 4 + 3, v.w * norm);
    }
}

// ---------------------------------------------------------------------------
// 5) h = relu(accum + b_conv); logits = h @ W_fc + b_fc; log_softmax
// ---------------------------------------------------------------------------
__global__ void head_kernel(const float* __restrict__ accum,
                            const float* __restrict__ b_conv,
                            const float* __restrict__ W_fc,
                            const float* __restrict__ b_fc,
                            float* __restrict__ out) {
    int i = blockIdx.x * blockDim.x + threadIdx.x;
    if (i >= N_NODES) return;
    float h[D];
#pragma unroll
    for (int k = 0; k < D; ++k) {
        float v = accum[(size_t)i * D + k] + b_conv[k];
        h[k] = v > 0.0f ? v : 0.0f;
    }
    float l[N_CLS];
#pragma unroll
    for (int c = 0; c < N_CLS; ++c) {
        float s = b_fc[c];
#pragma unroll
        for (int k = 0; k < D; ++k) s = fmaf(h[k], W_fc[k * N_CLS + c], s);
        l[c] = s;
    }
    float m = l[0];
#pragma unroll
    for (int c = 1; c < N_CLS; ++c) m = fmaxf(m, l[c]);
    float sum = 0.0f;
#pragma unroll
    for (int c = 0; c < N_CLS; ++c) sum += __expf(l[c] - m);
    float lse = __logf(sum);
#pragma unroll
    for (int c = 0; c < N_CLS; ++c) out[(size_t)i * N_CLS + c] = l[c] - m - lse;
}

// ---------------------------------------------------------------------------
extern "C" void kernel_launch(void* const* d_in, const int* in_sizes, int n_in,
                              void* d_out, int out_size, void* d_ws, size_t ws_size,
                              hipStream_t stream) {
    const float*     x      = (const float*)d_in[0];
    const long long* eidx   = (const long long*)d_in[1];   // int64 per reference
    const float*     W_conv = (const float*)d_in[2];
    const float*     b_conv = (const float*)d_in[3];
    const float*     W_fc   = (const float*)d_in[4];
    const float*     b_fc   = (const float*)d_in[5];
    float*           out    = (float*)d_out;

    const long long* row = eidx;             // edge_index[0] : source
    const long long* col = eidx + N_EDGES;   // edge_index[1] : target

    char* ws = (char*)d_ws;
    float* dinv  = (float*)(ws);                                   //  2.0 MB
    float* xw    = (float*)(ws + (size_t)N_NODES * 4);             // 32.0 MB
    float* accum = (float*)(ws + (size_t)N_NODES * 4
                               + (size_t)N_NODES * D * 4);         // 32.0 MB

    const int B = 256;
    int gN = (N_NODES + B - 1) / B;
    int gE = (N_EDGES + B - 1) / B;
    int nWaves = N_NODES / 16;                       // 31250 tiles
    int gW = (nWaves * 32 + B - 1) / B;              // 8 waves per block

    init_deg_kernel<<<gN, B, 0, stream>>>(dinv);
    deg_accum_kernel<<<gE, B, 0, stream>>>(col, dinv);
    dinv_kernel<<<gN, B, 0, stream>>>(dinv);
    xw_wmma_kernel<<<gW, B, 0, stream>>>(x, W_conv, xw);
    init_accum_kernel<<<gN, B, 0, stream>>>(xw, dinv, accum);
    scatter_kernel<<<gE, B, 0, stream>>>(row, col, xw, dinv, accum);
    head_kernel<<<gN, B, 0, stream>>>(accum, b_conv, W_fc, b_fc, out);
}